// convPbc_1d_21973052686653
// MI455X (gfx1250) — compile-verified
//
#include <hip/hip_runtime.h>

// CDNA5 / gfx1250: wave32, WMMA f32 16x16x4.
typedef __attribute__((ext_vector_type(2))) float v2f;
typedef __attribute__((ext_vector_type(8))) float v8f;

#define C_IN   8
#define C_OUT  8
#define FSZ    4
#define TILE   16           // output positions per WMMA tile (N dimension)
#define TILES_PER_WAVE 16   // 256 output positions per wave
#define WAVES_PER_BLOCK 8   // 256 threads

// Fixed problem size from the reference; compile-time so channel/row strides
// fold into the 24-bit instruction offsets and wrap becomes an AND.
constexpr int kL = 65536;

__global__ __launch_bounds__(256)
void convPbc_wmma_kernel(const float* __restrict__ x,
                         const float* __restrict__ W,
                         const float* __restrict__ bias,
                         float* __restrict__ out) {
  const int lane = threadIdx.x & 31;
  const int wave = threadIdx.x >> 5;
  const int m    = lane & 15;   // row (A / C,D) or column n (B)
  const int hi   = lane >> 4;   // half-wave selector

  constexpr int posPerBlock    = WAVES_PER_BLOCK * TILES_PER_WAVE * TILE; // 2048
  constexpr int blocksPerBatch = kL / posPerBlock;                        // 32
  const int b       = blockIdx.x / blocksPerBatch;
  const int lblk    = (blockIdx.x % blocksPerBatch) * posPerBlock;
  const int wave_l0 = lblk + wave * (TILES_PER_WAVE * TILE);

  // ---- A fragments: one K=4 step per input channel c ----
  // A_c[row, kk] = W[row, c, kk] for row < C_OUT, else 0 (pad to M=16).
  // 32-bit A layout: VGPR0 holds K=0 (lanes 0-15) / K=2 (lanes 16-31),
  //                  VGPR1 holds K=1 / K=3.
  v2f A[C_IN];
  #pragma unroll
  for (int c = 0; c < C_IN; ++c) {
    float a0 = 0.0f, a1 = 0.0f;
    if (m < C_OUT) {
      const float* wp = W + (m * C_IN + c) * FSZ;
      a0 = wp[hi ? 2 : 0];
      a1 = wp[hi ? 3 : 1];
    }
    A[c].x = a0;
    A[c].y = a1;
  }

  // Bias in C/D layout: VGPR r holds row (r + hi*8); rows >= 8 are padding.
  v8f cbias;
  #pragma unroll
  for (int r = 0; r < 8; ++r) cbias[r] = hi ? 0.0f : bias[r];

  const float* xb = x   + (long long)b * C_IN  * kL;
  float*       ob = out + (long long)b * C_OUT * kL;
  const int n    = m;            // position within the 16-wide tile
  const int off0 = n + hi * 2;   // this half-wave holds taps {2*hi, 2*hi+1}
  const int off1 = off0 + 1;

  for (int t = 0; t < TILES_PER_WAVE; ++t) {
    const int l0 = wave_l0 + t * TILE;
    v8f acc = cbias;

    // Circular (PBC) wrap: L is a power of two -> single AND per tap.
    const int i0 = (l0 + off0) & (kL - 1);
    const int i1 = (l0 + off1) & (kL - 1);
    const float* p0 = xb + i0;   // tap koff   base (per-lane)
    const float* p1 = xb + i1;   // tap koff+1 base (per-lane)

    #pragma unroll
    for (int c = 0; c < C_IN; ++c) {
      v2f Bf;
      Bf.x = p0[c * kL];   // global_load_b32 ..., offset:c*256KB
      Bf.y = p1[c * kL];
      // acc += A_c (16x4) * B_c (4x16)
      acc = __builtin_amdgcn_wmma_f32_16x16x4_f32(
          /*neg_a=*/false, A[c], /*neg_b=*/false, Bf,
          /*c_mod=*/(short)0, acc, /*reuse_a=*/false, /*reuse_b=*/false);
    }

    // Rows 0..7 live in lanes 0-15 (one row per acc VGPR). Coalesced,
    // streaming (non-temporal) stores: output is write-once, keep L2 for x.
    if (hi == 0) {
      float* q = ob + l0 + n;
      #pragma unroll
      for (int r = 0; r < C_OUT; ++r)
        __builtin_nontemporal_store(acc[r], q + r * kL);
    }
  }
}

extern "C" void kernel_launch(void* const* d_in, const int* in_sizes, int n_in,
                              void* d_out, int out_size, void* d_ws, size_t ws_size,
                              hipStream_t stream) {
  const float* x    = (const float*)d_in[0];  // (B, C_in, L) fp32
  const float* W    = (const float*)d_in[1];  // (C_out, C_in, 4) fp32
  const float* bias = (const float*)d_in[2];  // (C_out,) fp32
  float*       out  = (float*)d_out;          // (B, C_out, L) fp32

  const int B = in_sizes[0] / (C_IN * kL);    // = 64

  constexpr int posPerBlock = WAVES_PER_BLOCK * TILES_PER_WAVE * TILE; // 2048
  const int blocks = B * (kL / posPerBlock);  // 64 * 32 = 2048 blocks

  convPbc_wmma_kernel<<<blocks, 256, 0, stream>>>(x, W, bias, out);
}